// GravNetLayer_31250182046110
// MI455X (gfx1250) — compile-verified
//
#include <hip/hip_runtime.h>
#include <cmath>

typedef __attribute__((ext_vector_type(2))) float v2f;
typedef __attribute__((ext_vector_type(8))) float v8f;

#define N_PTS   12000
#define DIN     64
#define DLAT    16
#define DFEAT   64
#define KNN     16
#define ROW_TILES (N_PTS / 16)               // 750 (N divisible by 16)
#define WAVES_PER_WG 8
#define NUM_WG  ((ROW_TILES + WAVES_PER_WG - 1) / WAVES_PER_WG)  // 94

static __device__ __forceinline__ v8f wmma_f32(v2f a, v2f b, v8f c) {
  // D = A(16x4 f32) * B(4x16 f32) + C(16x16 f32)
  return __builtin_amdgcn_wmma_f32_16x16x4_f32(
      /*neg_a=*/false, a, /*neg_b=*/false, b,
      /*c_mod=*/(short)0, c, /*reuse_a=*/false, /*reuse_b=*/false);
}

// ---------------------------------------------------------------------------
// Kernel 1: feats = tanh(x@W_feat + b_feat), coords = tanh(x@W_lat + b_lat)
// One wave per 16-row tile; x tile lives in WMMA A-registers (16 K4-slices).
// ---------------------------------------------------------------------------
__global__ void __launch_bounds__(256)
encoder_kernel(const float* __restrict__ x,
               const float* __restrict__ Wf, const float* __restrict__ bf,
               const float* __restrict__ Wl, const float* __restrict__ bl,
               float* __restrict__ feats, float* __restrict__ coords) {
  const int lane = threadIdx.x & 31;
  const int wave = threadIdx.x >> 5;
  const int tile = blockIdx.x * WAVES_PER_WG + wave;
  if (tile >= ROW_TILES) return;
  const int row0 = tile * 16;
  const int m  = lane & 15;     // A: row M, B/C: col N
  const int kh = lane >> 4;     // K-half select

  // A-slices of x[row0..row0+15][0..63]: slice s covers K = 4s..4s+3
  v2f a[16];
  const float* xr = x + (size_t)(row0 + m) * DIN;
  #pragma unroll
  for (int s = 0; s < 16; ++s) {
    a[s].x = xr[4 * s + 2 * kh];
    a[s].y = xr[4 * s + 1 + 2 * kh];
  }

  // feats: 4 column tiles of 16, accumulate over K=64 (16 WMMAs each)
  #pragma unroll
  for (int nt = 0; nt < 4; ++nt) {
    v8f acc = {};
    #pragma unroll
    for (int s = 0; s < 16; ++s) {
      v2f b;
      b.x = Wf[(4 * s + 2 * kh) * DFEAT + nt * 16 + m];
      b.y = Wf[(4 * s + 1 + 2 * kh) * DFEAT + nt * 16 + m];
      acc = wmma_f32(a[s], b, acc);
    }
    #pragma unroll
    for (int r = 0; r < 8; ++r) {
      const int row = row0 + r + 8 * kh;
      const int col = nt * 16 + m;
      feats[(size_t)row * DFEAT + col] = tanhf(acc[r] + bf[col]);
    }
  }

  // coords: one 16-col tile
  {
    v8f acc = {};
    #pragma unroll
    for (int s = 0; s < 16; ++s) {
      v2f b;
      b.x = Wl[(4 * s + 2 * kh) * DLAT + m];
      b.y = Wl[(4 * s + 1 + 2 * kh) * DLAT + m];
      acc = wmma_f32(a[s], b, acc);
    }
    #pragma unroll
    for (int r = 0; r < 8; ++r) {
      const int row = row0 + r + 8 * kh;
      coords[(size_t)row * DLAT + m] = tanhf(acc[r] + bl[m]);
    }
  }
}

// ---------------------------------------------------------------------------
// Kernel 2: sq[i] = ||coords_i||^2
// ---------------------------------------------------------------------------
__global__ void sq_kernel(const float* __restrict__ coords, float* __restrict__ sq) {
  const int i = blockIdx.x * blockDim.x + threadIdx.x;
  if (i >= N_PTS) return;
  const float* c = coords + (size_t)i * DLAT;
  float s = 0.f;
  #pragma unroll
  for (int d = 0; d < DLAT; ++d) s += c[d] * c[d];
  sq[i] = s;
}

// ---------------------------------------------------------------------------
// Kernel 3 (fused): tiled pairwise distances via f32 WMMA + per-row top-16
// (never materializes NxN), then w=exp(-d2) aggregation, then the final
// [feats|agg] @ W_out GEMM (K=128) also via WMMA.
// One wave owns a 16-row tile; 8 waves / 256-thread workgroup.
// ---------------------------------------------------------------------------
__global__ void __launch_bounds__(256)
gravnet_kernel(const float* __restrict__ coords, const float* __restrict__ sq,
               const float* __restrict__ feats,
               const float* __restrict__ Wout, const float* __restrict__ bout,
               float* __restrict__ out) {
  __shared__ float lds_dot[WAVES_PER_WG][16 * 17];          // dot tile, padded
  __shared__ float lds_w  [WAVES_PER_WG][16][KNN];          // exp(-d2) weights
  __shared__ int   lds_j  [WAVES_PER_WG][16][KNN];          // neighbor indices
  __shared__ float lds_agg[WAVES_PER_WG][16][DFEAT];        // aggregated feats

  const int lane = threadIdx.x & 31;
  const int wave = threadIdx.x >> 5;
  const int tile = blockIdx.x * WAVES_PER_WG + wave;
  if (tile >= ROW_TILES) return;
  const int row0 = tile * 16;
  const int m  = lane & 15;
  const int kh = lane >> 4;

  // A-slices: my 16 coords rows (K = 16 -> 4 slices of K4)
  v2f a[4];
  {
    const float* cr = coords + (size_t)(row0 + m) * DLAT;
    #pragma unroll
    for (int s = 0; s < 4; ++s) {
      a[s].x = cr[4 * s + 2 * kh];
      a[s].y = cr[4 * s + 1 + 2 * kh];
    }
  }

  // Private top-K state: lane l (<16) owns row row0+l. Registers only.
  float bestd[KNN]; int bestj[KNN];
  #pragma unroll
  for (int s = 0; s < KNN; ++s) { bestd[s] = 3.4e38f; bestj[s] = 0; }
  float worst = 3.4e38f; int wslot = 0;
  const float sqi = sq[row0 + m];

  for (int jt = 0; jt < ROW_TILES; ++jt) {
    const int j0 = jt * 16;
    if (jt + 1 < ROW_TILES)  // hint next column tile into cache
      __builtin_prefetch(coords + (size_t)(j0 + 16 + m) * DLAT, 0, 3);

    // B-slices: B[k][n] = coords[j0+n][k]  (transposed on the fly)
    v8f acc = {};
    const float* cj = coords + (size_t)(j0 + m) * DLAT;
    #pragma unroll
    for (int s = 0; s < 4; ++s) {
      v2f b;
      b.x = cj[4 * s + 2 * kh];
      b.y = cj[4 * s + 1 + 2 * kh];
      acc = wmma_f32(a[s], b, acc);   // dot(coords_i, coords_j)
    }

    // Scatter dot tile to this wave's LDS region (row-major per row owner)
    #pragma unroll
    for (int r = 0; r < 8; ++r)
      lds_dot[wave][(r + 8 * kh) * 17 + m] = acc[r];
    __builtin_amdgcn_wave_barrier();

    if (lane < 16) {                    // lane owns one row's top-K
      const int row = row0 + lane;
      #pragma unroll
      for (int n = 0; n < 16; ++n) {
        const int j = j0 + n;
        float d = sqi + sq[j] - 2.f * lds_dot[wave][lane * 17 + n];
        d = fmaxf(d, 0.f);
        if (j != row && d < worst) {
          #pragma unroll
          for (int s = 0; s < KNN; ++s)          // dynamic slot -> cndmask chain
            if (s == wslot) { bestd[s] = d; bestj[s] = j; }
          worst = bestd[0]; wslot = 0;
          #pragma unroll
          for (int s = 1; s < KNN; ++s)
            if (bestd[s] > worst) { worst = bestd[s]; wslot = s; }
        }
      }
    }
    __builtin_amdgcn_wave_barrier();
  }

  // Publish (w, j) lists
  if (lane < 16) {
    #pragma unroll
    for (int k = 0; k < KNN; ++k) {
      lds_w[wave][lane][k] = expf(-bestd[k]);
      lds_j[wave][lane][k] = bestj[k];
    }
  }
  __builtin_amdgcn_wave_barrier();

  // Wave-cooperative aggregation: agg[r][d] = sum_k w_k * feats[j_k][d]
  for (int r = 0; r < 16; ++r) {
    float acc0 = 0.f, acc1 = 0.f;
    #pragma unroll
    for (int k = 0; k < KNN; ++k) {
      const int   j  = lds_j[wave][r][k];
      const float wk = lds_w[wave][r][k];
      acc0 += wk * feats[(size_t)j * DFEAT + lane];
      acc1 += wk * feats[(size_t)j * DFEAT + lane + 32];
    }
    lds_agg[wave][r][lane]      = acc0;
    lds_agg[wave][r][lane + 32] = acc1;
  }
  __builtin_amdgcn_wave_barrier();

  // Final GEMM: out[16][64] = [feats_tile | agg_tile] (16x128) @ W_out (128x64)
  const float* fr = feats + (size_t)(row0 + m) * DFEAT;
  #pragma unroll
  for (int nt = 0; nt < 4; ++nt) {
    v8f acc = {};
    #pragma unroll
    for (int s = 0; s < 16; ++s) {      // K = 0..63 from feats
      v2f av, bv;
      av.x = fr[4 * s + 2 * kh];
      av.y = fr[4 * s + 1 + 2 * kh];
      bv.x = Wout[(4 * s + 2 * kh) * DFEAT + nt * 16 + m];
      bv.y = Wout[(4 * s + 1 + 2 * kh) * DFEAT + nt * 16 + m];
      acc = wmma_f32(av, bv, acc);
    }
    #pragma unroll
    for (int s = 0; s < 16; ++s) {      // K = 64..127 from agg (LDS)
      v2f av, bv;
      av.x = lds_agg[wave][m][4 * s + 2 * kh];
      av.y = lds_agg[wave][m][4 * s + 1 + 2 * kh];
      bv.x = Wout[(64 + 4 * s + 2 * kh) * DFEAT + nt * 16 + m];
      bv.y = Wout[(64 + 4 * s + 1 + 2 * kh) * DFEAT + nt * 16 + m];
      acc = wmma_f32(av, bv, acc);
    }
    #pragma unroll
    for (int r = 0; r < 8; ++r) {
      const int row = row0 + r + 8 * kh;
      const int col = nt * 16 + m;
      out[(size_t)row * DFEAT + col] = acc[r] + bout[col];
    }
  }
}

// ---------------------------------------------------------------------------
extern "C" void kernel_launch(void* const* d_in, const int* in_sizes, int n_in,
                              void* d_out, int out_size, void* d_ws, size_t ws_size,
                              hipStream_t stream) {
  const float* x  = (const float*)d_in[0];
  const float* Wf = (const float*)d_in[1];
  const float* bf = (const float*)d_in[2];
  const float* Wl = (const float*)d_in[3];
  const float* bl = (const float*)d_in[4];
  const float* Wo = (const float*)d_in[5];
  const float* bo = (const float*)d_in[6];
  float* out = (float*)d_out;

  // Workspace layout: feats [N*64] | coords [N*16] | sq [N]
  float* feats  = (float*)d_ws;
  float* coords = feats  + (size_t)N_PTS * DFEAT;
  float* sq     = coords + (size_t)N_PTS * DLAT;

  encoder_kernel<<<NUM_WG, 256, 0, stream>>>(x, Wf, bf, Wl, bl, feats, coords);
  sq_kernel<<<(N_PTS + 255) / 256, 256, 0, stream>>>(coords, sq);
  gravnet_kernel<<<NUM_WG, 256, 0, stream>>>(coords, sq, feats, Wo, bo, out);
}